// TwoLayerGAT_32581621908235
// MI455X (gfx1250) — compile-verified
//
#include <hip/hip_runtime.h>

// ---------------------------------------------------------------------------
// Two-layer dense GAT for gfx1250 (MI455X), fp32 WMMA (V_WMMA_F32_16X16X4_F32).
// Roofline: dominated by 6144x6144 mask traffic -> pack to 1 bit/edge once
// (4.7 MB, L2-resident across all 10 attention passes). f32 WMMA keeps exact
// reference numerics at no cost since we are L2/HBM bound, not compute bound.
// All LDS operand panels are stored k-innermost with EVEN stride so every
// WMMA A/B fragment is one aligned, bank-conflict-free ds_load_b64 directly
// into the operand register pair (no v_mov shuffles before v_wmma).
// Workspace requirement: ~43 MB (see offsets in kernel_launch).
// ---------------------------------------------------------------------------

#define N_NODES 6144
#define ALPHA_LRELU 0.2f
#define NEGF -9.0e15f

typedef __attribute__((ext_vector_type(2))) float v2f;
typedef __attribute__((ext_vector_type(8))) float v8f;

// D = A(16x4,f32) * B(4x16,f32) + C(16x16,f32)
__device__ __forceinline__ v8f wmma4(v2f a, v2f b, v8f c) {
  return __builtin_amdgcn_wmma_f32_16x16x4_f32(
      /*neg_a=*/false, a, /*neg_b=*/false, b,
      /*c_mod=*/(short)0, c, /*reuse_a=*/false, /*reuse_b=*/false);
}

// ---------------------------------------------------------------------------
// 1) Pack adjacency int32 -> bitmask (1 bit/edge). Fully coalesced: each lane
//    reads one int, wave32 ballot produces the 32-bit word.
// ---------------------------------------------------------------------------
__global__ void pack_mask_kernel(const int* __restrict__ adj,
                                 unsigned* __restrict__ bm, long long total) {
  long long i = (long long)blockIdx.x * blockDim.x + threadIdx.x;
  int v = (i < total) ? adj[i] : 0;
  unsigned long long m = __ballot(v > 0); // wave32: low 32 bits valid
  if (((threadIdx.x & 31) == 0) && i < total) bm[i >> 5] = (unsigned)m;
}

// ---------------------------------------------------------------------------
// 2) Tiled WMMA GEMM: C[M,N] = act(A[M,K] @ B[K,N] (+bias)).
//    Block = 128 threads (4 waves), 64x64 output tile, KT=32 LDS staging.
//    A panel [row][k] and B panel [col][k], both stride 34 (even, coprime-ish
//    with 64 banks) -> every fragment is one aligned ds_load_b64.
//    Requires M % 64 == 0 and K % 32 == 0 (true for all uses here).
// ---------------------------------------------------------------------------
template <int ACT> // 0 = none, 1 = bias + relu
__global__ void gemm_wmma_kernel(const float* __restrict__ A,
                                 const float* __restrict__ B,
                                 const float* __restrict__ bias,
                                 float* __restrict__ C,
                                 int M, int K, int N, int lda, int ldb, int ldc) {
  __shared__ __align__(16) float As[64][34]; // [row][k]
  __shared__ __align__(16) float Bs[64][34]; // [col][k]  (k innermost!)

  const int tid  = threadIdx.x;
  const int w    = tid >> 5;
  const int lane = tid & 31;
  const int l15  = lane & 15;
  const int koff = (lane >> 4) * 2; // K split across lane halves (ISA layout)

  const int n0 = blockIdx.x * 64;
  const int m0 = blockIdx.y * 64;

  v8f acc[4];
#pragma unroll
  for (int t = 0; t < 4; t++)
#pragma unroll
    for (int g = 0; g < 8; g++) acc[t][g] = 0.0f;

  for (int k0 = 0; k0 < K; k0 += 32) {
    // Stage A tile 64x32 (coalesced: 32 lanes read 32 contiguous k of a row)
#pragma unroll
    for (int i = 0; i < 16; i++) {
      int idx = tid + i * 128;
      int r = idx >> 5, kk = idx & 31;
      As[r][kk] = A[(size_t)(m0 + r) * lda + (k0 + kk)];
    }
    // Stage B tile 32x64 transposed into [col][k] (coalesced along columns)
#pragma unroll
    for (int i = 0; i < 16; i++) {
      int idx = tid + i * 128;
      int c = idx & 63, kr = idx >> 6;
      int col = n0 + c;
      Bs[c][kr] = (col < N) ? B[(size_t)(k0 + kr) * ldb + col] : 0.0f;
    }
    __syncthreads();

#pragma unroll
    for (int k4 = 0; k4 < 8; k4++) {
      v2f a = *(const v2f*)&As[w * 16 + l15][k4 * 4 + koff]; // ds_load_b64
#pragma unroll
      for (int nt = 0; nt < 4; nt++) {
        v2f b = *(const v2f*)&Bs[nt * 16 + l15][k4 * 4 + koff]; // ds_load_b64
        acc[nt] = wmma4(a, b, acc[nt]); // v_wmma_f32_16x16x4_f32
      }
    }
    __syncthreads();
  }

  // Epilogue. C VGPR g: lanes 0-15 -> row g, lanes 16-31 -> row g+8.
#pragma unroll
  for (int nt = 0; nt < 4; nt++) {
#pragma unroll
    for (int g = 0; g < 8; g++) {
      int r = m0 + w * 16 + ((lane < 16) ? g : g + 8);
      int c = n0 + nt * 16 + l15;
      if (c < N) {
        float v = acc[nt][g];
        if (ACT == 1) {
          v += bias[c];
          v = v > 0.0f ? v : 0.0f;
        }
        C[(size_t)r * ldc + c] = v;
      }
    }
  }
}

// ---------------------------------------------------------------------------
// 3) f1[i] = Wh[i,:] . a1 ; f2[i] = Wh[i,:] . a2.  One wave per row,
//    __shfl_xor wave32 reduction.
// ---------------------------------------------------------------------------
__global__ void rowdot2_kernel(const float* __restrict__ Wh,
                               const float* __restrict__ a1,
                               const float* __restrict__ a2,
                               float* __restrict__ f1, float* __restrict__ f2,
                               int Fp, int ldw) {
  const int w = threadIdx.x >> 5, lane = threadIdx.x & 31;
  const int row = blockIdx.x * 4 + w;
  float s1 = 0.0f, s2 = 0.0f;
  for (int k = lane; k < Fp; k += 32) {
    float v = Wh[(size_t)row * ldw + k];
    s1 += v * a1[k];
    s2 += v * a2[k];
  }
#pragma unroll
  for (int off = 16; off >= 1; off >>= 1) {
    s1 += __shfl_xor(s1, off, 32);
    s2 += __shfl_xor(s2, off, 32);
  }
  if (lane == 0) {
    f1[row] = s1;
    f2[row] = s2;
  }
}

// ---------------------------------------------------------------------------
// 4) Row softmax stats: m[i] = max_j masked(e_ij), l[i] = sum_j exp(e-m).
//    One 256-thread block per row, online (m,l) accumulation + tree combine.
//    Mask read from the packed bitmask (L2-resident).
// ---------------------------------------------------------------------------
__device__ __forceinline__ void ol_combine(float& m, float& l, float m2, float l2) {
  float mn = fmaxf(m, m2);
  l = l * __expf(m - mn) + l2 * __expf(m2 - mn);
  m = mn;
}

__global__ void att_stats_kernel(const float* __restrict__ f1,
                                 const float* __restrict__ f2,
                                 const unsigned* __restrict__ bm,
                                 float* __restrict__ mrow, float* __restrict__ lrow) {
  const int row = blockIdx.x;
  const float f1r = f1[row];
  const unsigned* bmr = bm + (size_t)row * (N_NODES >> 5);

  float m = -3.0e38f, l = 0.0f;
  for (int j = threadIdx.x; j < N_NODES; j += 256) {
    float v = f1r + f2[j];
    float e = v > 0.0f ? v : ALPHA_LRELU * v;
    unsigned word = bmr[j >> 5]; // same word for the whole wave -> broadcast
    e = ((word >> (j & 31)) & 1u) ? e : NEGF;
    float mn = fmaxf(m, e);
    l = l * __expf(m - mn) + __expf(e - mn);
    m = mn;
  }
  // wave32 reduction
#pragma unroll
  for (int off = 16; off >= 1; off >>= 1) {
    float m2 = __shfl_xor(m, off, 32);
    float l2 = __shfl_xor(l, off, 32);
    ol_combine(m, l, m2, l2);
  }
  __shared__ float sm[8], sl[8];
  if ((threadIdx.x & 31) == 0) {
    sm[threadIdx.x >> 5] = m;
    sl[threadIdx.x >> 5] = l;
  }
  __syncthreads();
  if (threadIdx.x == 0) {
    float mm = sm[0], ll = sl[0];
#pragma unroll
    for (int i = 1; i < 8; i++) ol_combine(mm, ll, sm[i], sl[i]);
    mrow[row] = mm;
    lrow[row] = ll;
  }
}

// ---------------------------------------------------------------------------
// 5) Attention apply: out[32-row block, :FP] = act( softmax(e) @ Wh ).
//    Block = 128 threads (4 waves), 32 output rows (halves Wh/L2 re-reads vs
//    a 16-row block). Per 64-column tile the block cooperatively stages:
//      * p[32][stride66]   : normalized probabilities (two-pass softmax, so
//                            no accumulator rescale is ever needed)
//      * Whs[FP][stride66] : the 64xFP Wh panel, k-innermost (coalesced 128B
//                            global reads; B-fragments = one ds_load_b64)
//    Wave w owns row sub-block s = w&1 and feature tiles nt = (w>>1)+2t, so
//    all 4 waves are busy for every FP in {32,64,128,256} and every WMMA sits
//    under wave-uniform control flow (EXEC all-ones).
//    ACT: 0 = elu, 1 = sigmoid(elu).
// ---------------------------------------------------------------------------
template <int FP, int ACT>
__global__ void att_apply_kernel(const float* __restrict__ Wh, // [N, FP] dense
                                 const float* __restrict__ f1,
                                 const float* __restrict__ f2,
                                 const float* __restrict__ mrow,
                                 const float* __restrict__ lrow,
                                 const unsigned* __restrict__ bm,
                                 float* __restrict__ out, int ldc) {
  constexpr int NT = FP / 16;          // 16-wide feature tiles
  constexpr int TPW = (2 * NT) / 4;    // (s,nt) instances per wave, >= 1
  constexpr int LS = 66;               // even LDS stride -> aligned b64, no conflicts

  __shared__ __align__(16) float pt[32 * LS];       // p tile [row][k]
  __shared__ __align__(16) float whs[FP * LS];      // Wh panel [feat][k]

  const int rb = blockIdx.x; // 32-row block
  const int tid = threadIdx.x;
  const int w = tid >> 5, lane = tid & 31;
  const int l15 = lane & 15;
  const int koff = (lane >> 4) * 2;
  const int s = w & 1;       // row sub-block owned by this wave (uniform)

  v8f acc[TPW];
#pragma unroll
  for (int t = 0; t < TPW; t++)
#pragma unroll
    for (int g = 0; g < 8; g++) acc[t][g] = 0.0f;

  // p-compute mapping: thread -> (row = tid&31, 16-column group = tid>>5)
  const int prow = tid & 31;
  const int pcg = tid >> 5;
  const int grow = rb * 32 + prow;
  const float f1r = f1[grow];
  const float mr = mrow[grow];
  const float il = 1.0f / lrow[grow];
  const unsigned* bmr = bm + (size_t)grow * (N_NODES >> 5);

  for (int jb = 0; jb < N_NODES; jb += 64) {
    if (jb + 64 < N_NODES) // emits global_prefetch_b8 for the next Wh panel
      __builtin_prefetch(Wh + (size_t)(jb + 64) * FP, 0, 1);

    // ---- stage Wh panel 64xFP, transposed to [feat][k] ------------------
    // Consecutive tids read consecutive features of one row: coalesced 128B.
#pragma unroll
    for (int i = tid; i < 64 * FP; i += 128) {
      int kr = i / FP;        // panel row   (compile-time shift: FP is 2^k)
      int n = i & (FP - 1);   // feature col
      whs[n * LS + kr] = Wh[(size_t)(jb + kr) * FP + n];
    }

    // ---- block-cooperative probability tile -----------------------------
    {
      unsigned word = bmr[(jb >> 5) + (pcg >> 1)];
      int cbase = jb + pcg * 16;
      float* prow_p = &pt[prow * LS + pcg * 16];
#pragma unroll
      for (int c = 0; c < 16; c++) {
        int col = cbase + c;
        float v = f1r + f2[col];
        float e = v > 0.0f ? v : ALPHA_LRELU * v;
        prow_p[c] = ((word >> (col & 31)) & 1u) ? __expf(e - mr) * il : 0.0f;
      }
    }
    __syncthreads();

    // ---- WMMA accumulate: p(32x64) @ Wh(64xFP), K-chunks of 4 -----------
#pragma unroll
    for (int kc = 0; kc < 16; kc++) {
      v2f a = *(const v2f*)&pt[(s * 16 + l15) * LS + kc * 4 + koff]; // b64
#pragma unroll
      for (int t = 0; t < TPW; t++) {
        int nt = (w >> 1) + 2 * t; // wave-uniform
        if (nt < NT) {
          v2f b = *(const v2f*)&whs[(nt * 16 + l15) * LS + kc * 4 + koff];
          acc[t] = wmma4(a, b, acc[t]); // v_wmma_f32_16x16x4_f32
        }
      }
    }
    __syncthreads();
  }

  // ---- epilogue: activation + store ------------------------------------
#pragma unroll
  for (int t = 0; t < TPW; t++) {
    int nt = (w >> 1) + 2 * t;
    if (nt < NT) {
#pragma unroll
      for (int g = 0; g < 8; g++) {
        int r = rb * 32 + s * 16 + ((lane < 16) ? g : g + 8);
        int c = nt * 16 + l15;
        float v = acc[t][g];
        v = v > 0.0f ? v : __expf(v) - 1.0f; // elu (alpha=1)
        if (ACT == 1) v = 1.0f / (1.0f + __expf(-v)); // sigmoid
        out[(size_t)r * ldc + c] = v;
      }
    }
  }
}

// ---------------------------------------------------------------------------
// Host-side orchestration (all launches on `stream`; graph-capture safe).
// ---------------------------------------------------------------------------
extern "C" void kernel_launch(void* const* d_in, const int* in_sizes, int n_in,
                              void* d_out, int out_size, void* d_ws, size_t ws_size,
                              hipStream_t stream) {
  const float* x   = (const float*)d_in[0];
  const int*   adj = (const int*)d_in[1];
  const float* W1  = (const float*)d_in[2]; // [4,256,64]
  const float* a1  = (const float*)d_in[3]; // [4,128]
  const float* Wo  = (const float*)d_in[4]; // [256,256]
  const float* ao  = (const float*)d_in[5]; // [512]
  const float* W2  = (const float*)d_in[6]; // [4,256,32]
  const float* a2  = (const float*)d_in[7]; // [4,64]
  const float* Wo2 = (const float*)d_in[8]; // [128,128]
  const float* ao2 = (const float*)d_in[9]; // [256]
  const float* lw  = (const float*)d_in[10]; // [128,379]
  const float* lb  = (const float*)d_in[11]; // [379]
  float* out = (float*)d_out;
  float* ws = (float*)d_ws;

  const int N = N_NODES;
  size_t o = 0;
  float* WH1  = ws + o; o += (size_t)4 * N * 64;  // per-head Wh, layer 1
  float* H1   = ws + o; o += (size_t)N * 256;     // concat(elu heads)
  float* WHO  = ws + o; o += (size_t)N * 256;     // out-att-1 Wh
  float* HO   = ws + o; o += (size_t)N * 256;     // sigmoid(elu(...))
  float* WH2  = ws + o; o += (size_t)4 * N * 32;
  float* H2   = ws + o; o += (size_t)N * 128;
  float* WHO2 = ws + o; o += (size_t)N * 128;
  float* HO2  = ws + o; o += (size_t)N * 128;
  float* F1   = ws + o; o += N;
  float* F2   = ws + o; o += N;
  float* MR   = ws + o; o += N;
  float* LR   = ws + o; o += N;
  unsigned* BM = (unsigned*)(ws + o); // N*N/32 words = 4.7 MB (L2-resident)

  const int MB = N / 64;  // 96 GEMM row-blocks
  const int RB = N / 32;  // 192 attention row-blocks

  // ---- pack adjacency once: 151 MB int32 -> 4.7 MB bits ------------------
  {
    long long total = (long long)N * N;
    pack_mask_kernel<<<(int)((total + 255) / 256), 256, 0, stream>>>(adj, BM, total);
  }

  // ---- layer 1: 4 heads, Fp = 64, concat + elu ---------------------------
  for (int h = 0; h < 4; h++) {
    gemm_wmma_kernel<0><<<dim3(1, MB), 128, 0, stream>>>(
        x, W1 + (size_t)h * 256 * 64, nullptr, WH1 + (size_t)h * N * 64,
        N, 256, 64, 256, 64, 64);
  }
  for (int h = 0; h < 4; h++) {
    const float* Whh = WH1 + (size_t)h * N * 64;
    rowdot2_kernel<<<N / 4, 128, 0, stream>>>(Whh, a1 + h * 128, a1 + h * 128 + 64,
                                              F1, F2, 64, 64);
    att_stats_kernel<<<N, 256, 0, stream>>>(F1, F2, BM, MR, LR);
    att_apply_kernel<64, 0><<<RB, 128, 0, stream>>>(Whh, F1, F2, MR, LR, BM,
                                                    H1 + h * 64, 256);
  }

  // ---- out attention 1: Fp = 256, sigmoid(elu) ---------------------------
  gemm_wmma_kernel<0><<<dim3(4, MB), 128, 0, stream>>>(H1, Wo, nullptr, WHO,
                                                       N, 256, 256, 256, 256, 256);
  rowdot2_kernel<<<N / 4, 128, 0, stream>>>(WHO, ao, ao + 256, F1, F2, 256, 256);
  att_stats_kernel<<<N, 256, 0, stream>>>(F1, F2, BM, MR, LR);
  att_apply_kernel<256, 1><<<RB, 128, 0, stream>>>(WHO, F1, F2, MR, LR, BM, HO, 256);

  // ---- layer 2: 4 heads, Fp = 32, concat + elu ---------------------------
  for (int h = 0; h < 4; h++) {
    gemm_wmma_kernel<0><<<dim3(1, MB), 128, 0, stream>>>(
        HO, W2 + (size_t)h * 256 * 32, nullptr, WH2 + (size_t)h * N * 32,
        N, 256, 32, 256, 32, 32);
  }
  for (int h = 0; h < 4; h++) {
    const float* Whh = WH2 + (size_t)h * N * 32;
    rowdot2_kernel<<<N / 4, 128, 0, stream>>>(Whh, a2 + h * 64, a2 + h * 64 + 32,
                                              F1, F2, 32, 32);
    att_stats_kernel<<<N, 256, 0, stream>>>(F1, F2, BM, MR, LR);
    att_apply_kernel<32, 0><<<RB, 128, 0, stream>>>(Whh, F1, F2, MR, LR, BM,
                                                    H2 + h * 32, 128);
  }

  // ---- out attention 2: Fp = 128, sigmoid(elu) ---------------------------
  gemm_wmma_kernel<0><<<dim3(2, MB), 128, 0, stream>>>(H2, Wo2, nullptr, WHO2,
                                                       N, 128, 128, 128, 128, 128);
  rowdot2_kernel<<<N / 4, 128, 0, stream>>>(WHO2, ao2, ao2 + 128, F1, F2, 128, 128);
  att_stats_kernel<<<N, 256, 0, stream>>>(F1, F2, BM, MR, LR);
  att_apply_kernel<128, 1><<<RB, 128, 0, stream>>>(WHO2, F1, F2, MR, LR, BM, HO2, 128);

  // ---- final linear + relu: [6144,128] @ [128,379] + b -------------------
  gemm_wmma_kernel<1><<<dim3((379 + 63) / 64, MB), 128, 0, stream>>>(
      HO2, lw, lb, out, N, 128, 379, 128, 379, 379);
}